// Attention_69544110457499
// MI455X (gfx1250) — compile-verified
//
#include <hip/hip_runtime.h>

typedef float v2f __attribute__((ext_vector_type(2)));
typedef float v8f __attribute__((ext_vector_type(8)));
typedef unsigned int v4u __attribute__((ext_vector_type(4)));
typedef int v4i __attribute__((ext_vector_type(4)));
typedef int v8i __attribute__((ext_vector_type(8)));

namespace {
constexpr int kH         = 12;
constexpr int kD         = 64;
constexpr int kBlockRows = 2048;
constexpr int kWindow    = 4096;
constexpr int kWaves     = 4;
constexpr int kRowsPerWave = kBlockRows / kWaves;  // 512
constexpr int kTile      = 16;
constexpr int kRowStride = kH * kD;                // 768 floats between cache rows
constexpr int kLdsPitch  = kD + 4;                 // 68 floats: conflict-free & TDM pad
constexpr int kBufFloats = kTile * kLdsPitch;      // 1088 floats per K tile buffer
}  // namespace

// Issue one TDM load of a 16x64 f32 tile (rows strided by 768 floats) into LDS
// at byte offset lds_off, padding +4 dwords after every 64 dwords (row pitch 68).
// D# layout per CDNA5 ISA (group0/group1; groups 2-3 unused).
__device__ __forceinline__ void tdm_load_k_tile(const float* gptr, unsigned lds_off) {
  const unsigned long long ga = (unsigned long long)(uintptr_t)gptr;
  v4u g0;
  g0[0] = 1u;                                           // count=1 valid descriptor
  g0[1] = lds_off;                                      // lds_addr (bytes)
  g0[2] = (unsigned)(ga & 0xFFFFFFFFu);                 // global_addr[31:0]
  g0[3] = (unsigned)((ga >> 32) & 0x1FFFFFFu)           // global_addr[56:32]
        | (2u << 30);                                   // type=2 ("image")
  v8i g1;
  g1[0] = (2 << 16)                                     // data_size = 4 bytes
        | (1 << 20)                                     // pad_enable
        | (5 << 22)                                     // pad_interval: 64 dwords
        | (3 << 25);                                    // pad_amount: 4 dwords
  g1[1] = (kD << 16);                                   // tensor_dim0 = 64
  g1[2] = (int)((unsigned)kBlockRows << 16);            // tensor_dim1 = 2048
  g1[3] = (kD << 16);                                   // tile_dim0 = 64
  g1[4] = kTile;                                        // tile_dim1 = 16 rows
  g1[5] = kRowStride;                                   // tensor_dim0_stride = 768
  g1[6] = 0;
  g1[7] = 0;
  const v4i z4 = {0, 0, 0, 0};                          // 2D tensor: groups 2/3 unused
  const v8i z8 = {0, 0, 0, 0, 0, 0, 0, 0};
  __builtin_amdgcn_tensor_load_to_lds(g0, g1, z4, z4, z8, 0);
}

__global__ __launch_bounds__(kWaves * 32)
void attn_kv_nosoftmax_kernel(const int* __restrict__ tptr,
                              const float* __restrict__ q,
                              const float* __restrict__ knew,
                              const float* __restrict__ vnew,
                              const float* __restrict__ kcache,
                              const float* __restrict__ vcache,
                              float* __restrict__ out)
{
  __shared__ __align__(16) float ldsK[kWaves][2][kBufFloats];  // double-buffered K tiles
  __shared__ float lds_part[kWaves * kD];  // per-wave partial outputs
  __shared__ float lds_pc[kD];             // q[d]*Kcache[t][d] products
  __shared__ float lds_pn[kD];             // q[d]*knew[d]      products

  const int tid  = threadIdx.x;
  const int wave = tid >> 5;
  const int lane = tid & 31;
  const int ln   = lane & 15;   // N index (0..15)
  const int hi   = lane >> 4;   // K half-select (0/1)

  const int n = blockIdx.x / kH;
  const int h = blockIdx.x % kH;

  const float* qp  = q    + ((long)n * kH + h) * kD;
  const float* kp  = knew + ((long)n * kH + h) * kD;
  const float* vp  = vnew + ((long)n * kH + h) * kD;
  const float* kcp = kcache + (long)n * kBlockRows * kRowStride + (long)h * kD;
  const float* vcp = vcache + (long)n * kBlockRows * kRowStride + (long)h * kD;

  int t = tptr[0] % kWindow;
  if (t > kBlockRows - 1) t = kBlockRows - 1;   // dynamic_update_slice clamp

  // Products feeding the two correction dot-products (row-t swap applied later).
  if (tid < kD) {
    const float qd = qp[tid];
    lds_pc[tid] = qd * kcp[(long)t * kRowStride + tid];
    lds_pn[tid] = qd * kp[tid];
  }

  // Wave-uniform (SGPR) values for TDM descriptors.
  const int bbase = wave * kRowsPerWave;
  const unsigned kbase_off = (unsigned)__builtin_amdgcn_readfirstlane(
      (int)(unsigned)(uintptr_t)&ldsK[wave][0][0]);
  constexpr unsigned kBufBytes = kBufFloats * 4;

  // Stage-1 A operand: q replicated across all 16 M rows.
  // A(16x4) f32 layout: lanes 0-15 hold K=0,1 in vgpr0,1; lanes 16-31 hold K=2,3.
  v2f qa[16];
  #pragma unroll
  for (int c = 0; c < 16; ++c)
    qa[c] = *(const v2f*)(qp + 4 * c + 2 * hi);

  // Persistent D accumulators for out[d], 4 groups of 16 d-columns.
  v8f acc[4] = {v8f{0}, v8f{0}, v8f{0}, v8f{0}};

  // Prologue: TDM-load tile 0 into buffer 0.
  tdm_load_k_tile(kcp + (long)bbase * kRowStride, kbase_off);

  for (int tb = 0; tb < kRowsPerWave; tb += kTile) {
    const int b0  = bbase + tb;
    const int buf = (tb >> 4) & 1;

    // Issue next tile's TDM into the other buffer, then wait for current tile.
    if (tb + kTile < kRowsPerWave) {
      tdm_load_k_tile(kcp + (long)(b0 + kTile) * kRowStride,
                      kbase_off + (unsigned)(buf ^ 1) * kBufBytes);
      __builtin_amdgcn_s_wait_tensorcnt(1);   // in-order: current tile is done
    } else {
      __builtin_amdgcn_s_wait_tensorcnt(0);
    }
    asm volatile("" ::: "memory");  // keep LDS reads after the tensorcnt wait

    // Prefetch next tile's V rows.
    if (tb + kTile < kRowsPerWave) {
      __builtin_prefetch(vcp + (long)(b0 + kTile + ln) * kRowStride + (long)hi * 32, 0, 0);
    }

    // ---- Stage 1: scores[b0+n] = q · Kcache[b0+n], K tile from LDS ----
    // B(4x16): lane ln holds column n=ln, rows K = 2*hi + {0,1}.
    // Pitch 68 => lanes (l, l+16) read banks 4*ln+4c+{0..3}: conflict-free.
    const float* pK = &ldsK[wave][buf][0];
    v8f sc = {0};
    #pragma unroll
    for (int c = 0; c < 16; ++c) {
      const v2f bk = *(const v2f*)(pK + kLdsPitch * ln + 4 * c + 2 * hi);
      sc = __builtin_amdgcn_wmma_f32_16x16x4_f32(false, qa[c], false, bk,
                                                 (short)0, sc, false, false);
    }
    // All M rows of A identical => sc[0] in lane l == score[b0 + (l&15)].
    const int s0 = __builtin_bit_cast(int, sc[0]);

    // ---- Stage 2: acc[g] += scores_tile x V_tile (V direct from global) ----
    #pragma unroll
    for (int j = 0; j < 4; ++j) {
      const int r = 4 * j + 2 * hi;  // base row within tile for this lane's K half
      v2f a2;
      a2[0] = __builtin_bit_cast(float, __builtin_amdgcn_ds_bpermute((r + 0) << 2, s0));
      a2[1] = __builtin_bit_cast(float, __builtin_amdgcn_ds_bpermute((r + 1) << 2, s0));
      const float* vrow0 = vcp + (long)(b0 + r) * kRowStride;
      const float* vrow1 = vrow0 + kRowStride;
      #pragma unroll
      for (int g = 0; g < 4; ++g) {
        v2f bv;
        bv[0] = vrow0[g * 16 + ln];   // coalesced: 16 lanes x 16 consecutive floats
        bv[1] = vrow1[g * 16 + ln];
        acc[g] = __builtin_amdgcn_wmma_f32_16x16x4_f32(false, a2, false, bv,
                                                       (short)0, acc[g], false, false);
      }
    }
  }

  // acc[g] vgpr0, lane l = out_partial[g*16 + (l&15)] (replicated across halves).
  if (lane < 16) {
    #pragma unroll
    for (int g = 0; g < 4; ++g)
      lds_part[wave * kD + g * 16 + ln] = acc[g][0];
  }
  __syncthreads();

  // Combine waves + apply row-t correction, write the 64 outputs.
  if (tid < kD) {
    float s = 0.f;
    #pragma unroll
    for (int w = 0; w < kWaves; ++w) s += lds_part[w * kD + tid];
    float sc_old = 0.f, sc_new = 0.f;
    for (int i = 0; i < kD; ++i) { sc_old += lds_pc[i]; sc_new += lds_pn[i]; }
    s += sc_new * vp[tid] - sc_old * vcp[(long)t * kRowStride + tid];
    out[((long)n * kH + h) * kD + tid] = s;
  }
}

extern "C" void kernel_launch(void* const* d_in, const int* in_sizes, int n_in,
                              void* d_out, int out_size, void* d_ws, size_t ws_size,
                              hipStream_t stream) {
  const int*   t  = (const int*)d_in[0];
  const float* q  = (const float*)d_in[1];
  const float* k  = (const float*)d_in[2];
  const float* v  = (const float*)d_in[3];
  const float* kc = (const float*)d_in[4];
  const float* vc = (const float*)d_in[5];
  float* out = (float*)d_out;

  const int S = in_sizes[1] / (kH * kD);  // 64 samples
  dim3 grid(S * kH);                      // one WG per (sample, head)
  dim3 block(kWaves * 32);                // 4 wave32s

  hipLaunchKernelGGL(attn_kv_nosoftmax_kernel, grid, block, 0, stream,
                     t, q, k, v, kc, vc, out);
}